// CustmConv_2757369004068
// MI455X (gfx1250) — compile-verified
//
#include <hip/hip_runtime.h>
#include <math.h>

typedef __attribute__((ext_vector_type(2))) float v2f;
typedef __attribute__((ext_vector_type(8))) float v8f;

#define HH 56
#define WW 56
#define DD 256
#define OUTC 256
#define HDIM 32
#define K2 25
#define TILE 8          // 8x8 pixel tile per workgroup
#define NPIX 64
#define HALO 12         // TILE + 2*pad
#define SPATIAL 144     // 12*12
#define CH_PER_STAGE 128  // 4 heads * 32 channels staged at a time
#define XS_FLOATS (CH_PER_STAGE * SPATIAL)   // 18432 floats = 72 KB
#define ATT_STRIDE 80                         // padded row stride (bank-conflict-free)
#define ATT_FLOATS (DD * ATT_STRIDE)          // 20480 floats = 80 KB
#define LDS_FLOATS (ATT_FLOATS + XS_FLOATS)   // 155648 bytes total

__global__ __launch_bounds__(256)
void natt_fused_kernel(const float* __restrict__ x,
                       const float* __restrict__ w_out,
                       const float* __restrict__ b_out,
                       float* __restrict__ out)
{
    extern __shared__ float lds[];
    float* att = lds;                 // [DD][ATT_STRIDE]
    float* xs  = lds + ATT_FLOATS;    // [CH_PER_STAGE][SPATIAL]

    const int t  = threadIdx.x;
    const int n  = blockIdx.y;
    const int th = blockIdx.x / 7;
    const int tw = blockIdx.x % 7;
    const int h0 = th * TILE;
    const int w0 = tw * TILE;

    const float* xn = x + (size_t)n * DD * HH * WW;

    const int hs = t >> 6;        // head-in-stage 0..3
    const int pl = t & 63;        // local pixel 0..63
    const int py = pl >> 3;
    const int px = pl & 7;

    // ---------------- Phase 1: local-window attention into LDS ----------------
    for (int it = 0; it < 2; ++it) {
        // stage 128 channels x 12x12 halo into LDS (zero-padded at borders)
        for (int idx = t; idx < XS_FLOATS; idx += 256) {
            int c  = idx / SPATIAL;
            int sp = idx - c * SPATIAL;
            int sy = sp / HALO;
            int sx = sp - sy * HALO;
            int gy = h0 - 2 + sy;
            int gx = w0 - 2 + sx;
            float v = 0.0f;
            if ((unsigned)gy < HH && (unsigned)gx < WW)
                v = xn[(size_t)(it * CH_PER_STAGE + c) * (HH * WW) + gy * WW + gx];
            xs[idx] = v;
        }
        __syncthreads();

        // one (pixel, head) task per thread
        const float* xh  = xs + hs * HDIM * SPATIAL;
        const int    spq = (py + 2) * HALO + (px + 2);

        float q[HDIM];
        #pragma unroll
        for (int c = 0; c < HDIM; ++c) q[c] = xh[c * SPATIAL + spq];

        float sc[K2];
        float m = -3.0e38f;
        #pragma unroll
        for (int j = 0; j < K2; ++j) {
            int dy = j / 5, dx = j - 5 * (j / 5);
            int sp = (py + dy) * HALO + (px + dx);
            float s = 0.0f;
            #pragma unroll
            for (int c = 0; c < HDIM; ++c) s += q[c] * xh[c * SPATIAL + sp];
            sc[j] = s;
            m = fmaxf(m, s);
        }
        float sum = 0.0f;
        #pragma unroll
        for (int j = 0; j < K2; ++j) { sc[j] = __expf(sc[j] - m); sum += sc[j]; }
        float inv = 1.0f / sum;

        float acc[HDIM];
        #pragma unroll
        for (int c = 0; c < HDIM; ++c) acc[c] = 0.0f;
        #pragma unroll
        for (int j = 0; j < K2; ++j) {
            int dy = j / 5, dx = j - 5 * (j / 5);
            int sp = (py + dy) * HALO + (px + dx);
            float wj = sc[j] * inv;
            #pragma unroll
            for (int c = 0; c < HDIM; ++c) acc[c] += wj * xh[c * SPATIAL + sp];
        }
        int cbase = (it * 4 + hs) * HDIM;
        #pragma unroll
        for (int c = 0; c < HDIM; ++c) att[(cbase + c) * ATT_STRIDE + pl] = acc[c];
        __syncthreads();   // protect xs before next stage overwrites it
    }

    // ---------------- Phase 2: 1x1 conv GEMM via V_WMMA_F32_16X16X4_F32 ------
    // out[o0:o0+16, p0:p0+16] per wave-tile; K=256 swept 4 at a time.
    const int wv   = t >> 5;
    const int lane = t & 31;
    const int lrow = lane & 15;
    const int lhi  = lane >> 4;
    float* outn = out + (size_t)n * OUTC * HH * WW;

    for (int tile = wv; tile < 64; tile += 8) {
        int o0 = (tile >> 2) * 16;
        int p0 = (tile & 3) * 16;

        v8f c8;
        #pragma unroll
        for (int i = 0; i < 8; ++i) c8[i] = b_out[o0 + i + 8 * lhi];  // bias-init C

        // A: lane holds W[o0+lrow, k + 2*lhi + {0,1}]  (ISA 16x4 f32 A layout)
        const float* wrow = w_out + (size_t)(o0 + lrow) * DD + 2 * lhi;
        for (int k = 0; k < DD; k += 4) {
            v2f a;
            a.x = wrow[k];
            a.y = wrow[k + 1];
            // B: lane holds att[k + 2*lhi + {0,1}][p0 + lrow]
            const float* bp = att + (size_t)(k + 2 * lhi) * ATT_STRIDE + p0 + lrow;
            v2f b;
            b.x = bp[0];
            b.y = bp[ATT_STRIDE];
            c8 = __builtin_amdgcn_wmma_f32_16x16x4_f32(
                /*neg_a=*/false, a, /*neg_b=*/false, b,
                /*c_mod=*/(short)0, c8, /*reuse_a=*/false, /*reuse_b=*/false);
        }

        // scatter 16x16 f32 C/D tile: VGPR i, lane -> row o0+i+8*lhi, col p0+lrow
        int plix = p0 + lrow;
        int gp = (h0 + (plix >> 3)) * WW + (w0 + (plix & 7));
        #pragma unroll
        for (int i = 0; i < 8; ++i)
            outn[(size_t)(o0 + i + 8 * lhi) * (HH * WW) + gp] = c8[i];
    }
}

extern "C" void kernel_launch(void* const* d_in, const int* in_sizes, int n_in,
                              void* d_out, int out_size, void* d_ws, size_t ws_size,
                              hipStream_t stream) {
    const float* x     = (const float*)d_in[0];
    const float* w_out = (const float*)d_in[1];
    const float* b_out = (const float*)d_in[2];
    float* out = (float*)d_out;

    int N = in_sizes[0] / (DD * HH * WW);   // 8
    dim3 grid(49, N);                        // 7x7 spatial tiles per image
    size_t shmem = (size_t)LDS_FLOATS * sizeof(float);
    natt_fused_kernel<<<grid, 256, shmem, stream>>>(x, w_out, b_out, out);
}